// Vanilla_317827580171
// MI455X (gfx1250) — compile-verified
//
#include <hip/hip_runtime.h>
#include <hip/hip_bf16.h>

// ---------------------------------------------------------------------------
// Fused QKV projection + KV-cache attention for MI455X (gfx1250, wave32).
// Memory-bound (~450MB f32 @ 23.3TB/s); all GEMMs use V_WMMA_F32_16X16X4_F32.
// Adds: global_prefetch_b8 on the W stream, TDM (tensor_load_to_lds) staging
// of the per-head Q tile with s_wait_tensorcnt.
// ---------------------------------------------------------------------------

typedef __attribute__((ext_vector_type(2))) float v2f;
typedef __attribute__((ext_vector_type(8))) float v8f;
typedef __attribute__((ext_vector_type(4))) unsigned int u32x4;
typedef __attribute__((ext_vector_type(8))) unsigned int u32x8;

constexpr int M_TOK   = 16;
constexpr int N_DIM   = 4096;
constexpr int D_HEAD  = 128;
constexpr int P_CACHE = 8192;
constexpr int H_HEADS = 32;            // N_DIM / D_HEAD
constexpr int S_TOTAL = P_CACHE + M_TOK;   // 8208 (= 16 * 513)
constexpr int CHUNK   = 512;
constexpr int NCHUNK  = (S_TOTAL + CHUNK - 1) / CHUNK;  // 17
constexpr int KSPLIT  = 4;             // K-split of the QKV GEMM
constexpr int QKV_ELEMS = 3 * H_HEADS * M_TOK * D_HEAD; // 196608 floats

__device__ __forceinline__ v8f wmma_f32_16x16x4(v2f a, v2f b, v8f c) {
    return __builtin_amdgcn_wmma_f32_16x16x4_f32(
        /*neg_a=*/false, a, /*neg_b=*/false, b,
        /*c_mod=*/(short)0, c, /*reuse_a=*/false, /*reuse_b=*/false);
}

// ---------------------------------------------------------------------------
// TDM: DMA a contiguous run of `elems` f32 from global to LDS (1-row 2D tile).
// D# per CDNA5 ISA 8.3/8.4: group0 = {count|flags, lds_addr, gaddr_lo,
// gaddr_hi|type=2}; group1 packs data_size=4B, tensor_dim0/tile_dim0 = elems,
// tensor_dim1/tile_dim1 = 1, tensor_dim0_stride = elems.
// 2D form: VADDR2/VADDR3 omitted (NULL).
// ---------------------------------------------------------------------------
__device__ __forceinline__ void tdm_load_row_f32(unsigned int lds_addr,
                                                 unsigned long long gaddr,
                                                 unsigned int elems) {
    u32x4 g0;
    g0.x = 1u;                                            // count=1, user D#
    g0.y = lds_addr;                                      // LDS byte address
    g0.z = (unsigned int)gaddr;                           // global_addr[31:0]
    g0.w = ((unsigned int)(gaddr >> 32) & 0x01FFFFFFu)    // global_addr[56:32]
           | (2u << 30);                                  // type=2 ("image")
    u32x8 g1;
    g1.s0 = 0x00020000u;                                  // data_size=2 -> 4B
    g1.s1 = (elems & 0xFFFFu) << 16;                      // tensor_dim0[15:0]
    g1.s2 = ((elems >> 16) & 0xFFFFu) | (1u << 16);       // dim0 hi | tensor_dim1=1
    g1.s3 = (elems & 0xFFFFu) << 16;                      // tile_dim0
    g1.s4 = 1u;                                           // tile_dim1=1
    g1.s5 = elems;                                        // dim0_stride[31:0]
    g1.s6 = 0u;
    g1.s7 = 0u;
    asm volatile("tensor_load_to_lds %0, %1" :: "s"(g0), "s"(g1) : "memory");
}

// ---------------------------------------------------------------------------
// Kernel 0: zero the Q/K_new/V_new accumulation buffers (atomics target).
// ---------------------------------------------------------------------------
__global__ __launch_bounds__(256) void zero_qkv(float* __restrict__ qkv) {
    for (int i = blockIdx.x * 256 + threadIdx.x; i < QKV_ELEMS; i += gridDim.x * 256)
        qkv[i] = 0.0f;
}

// ---------------------------------------------------------------------------
// Kernel 1: C[16,4096] = X[16,4096] @ W for W in {Wq,Wk,Wv}.
// One wave per 16x16 output tile per K-split (3 * 256 tiles * 4 splits = 3072
// waves). Results combined with global f32 atomics; written as [H, M, D].
// ---------------------------------------------------------------------------
__global__ __launch_bounds__(256) void qkv_gemm(
    const float* __restrict__ X,
    const float* __restrict__ Wq,
    const float* __restrict__ Wk,
    const float* __restrict__ Wv,
    float* __restrict__ qkv) {

    const int lane  = threadIdx.x & 31;
    const int job   = blockIdx.x * 8 + (threadIdx.x >> 5);  // 0..3071
    const int tile  = job >> 2;          // 0..767
    const int ks    = job & (KSPLIT - 1);
    const int mi    = tile >> 8;         // 0=Q,1=K,2=V
    const int ct    = tile & 255;        // 16-col tile of N

    const float* W = (mi == 0) ? Wq : ((mi == 1) ? Wk : Wv);
    float* out = qkv + mi * (H_HEADS * M_TOK * D_HEAD);

    const int row  = lane & 15;          // A fragment: M row
    const int half = lane >> 4;          // lane half selects K sub-pair
    const int col  = lane & 15;          // B fragment: N col
    const int n    = ct * 16 + col;

    const int k_begin = ks * (N_DIM / KSPLIT);
    const int k_end   = k_begin + (N_DIM / KSPLIT);

    v8f acc = {};
    #pragma unroll 4
    for (int k0 = k_begin; k0 < k_end; k0 += 4) {
        // Stream-ahead prefetch of the W column block (once per 16 K-values).
        if ((k0 & 15) == 0 && (k0 + 64) < k_end)
            __builtin_prefetch(&W[(size_t)(k0 + 64 + half * 2) * N_DIM + n], 0, 0);
        // A (16x4 f32): VGPR0 = K{0|2}, VGPR1 = K{1|3} per lane-half
        v2f a = *(const v2f*)(X + row * N_DIM + k0 + half * 2);
        // B (4x16 f32): mirrored layout, B[k][n] = W[k*N + n]
        v2f b;
        b.x = W[(size_t)(k0 + half * 2 + 0) * N_DIM + n];
        b.y = W[(size_t)(k0 + half * 2 + 1) * N_DIM + n];
        acc = wmma_f32_16x16x4(a, b, acc);
    }

    // C/D layout: VGPR j -> row (j + half*8), col = lane&15
    const int h = n / D_HEAD;
    const int d = n % D_HEAD;
    #pragma unroll
    for (int j = 0; j < 8; ++j) {
        int r = j + half * 8;
        atomicAdd(&out[((size_t)h * M_TOK + r) * D_HEAD + d], acc[j]);
    }
}

// ---------------------------------------------------------------------------
// Kernel 2: per-(head, chunk) flash-attention partial.
//   Q tile staged global->LDS by the Tensor Data Mover (wave 0 issues,
//   s_wait_tensorcnt, then block barrier).
//   scores[16, CHUNK] = Q_h @ K_chunk^T        (waves split chunk rows)
//   chunk softmax stats (m, l) + exp in LDS
//   O_part[16, 128]   = P @ V_chunk            (waves split D columns)
// ---------------------------------------------------------------------------
__global__ __launch_bounds__(256) void attn_chunk(
    const float* __restrict__ cacheK,
    const float* __restrict__ cacheV,
    const float* __restrict__ qkv,
    float* __restrict__ o_part,
    float* __restrict__ stat_m,
    float* __restrict__ stat_l) {

    __shared__ float Qs[M_TOK * D_HEAD];     // 8 KB
    __shared__ float Ss[M_TOK][CHUNK];       // 32 KB
    __shared__ float red[M_TOK][16];         // 1 KB

    const int h  = blockIdx.x / NCHUNK;
    const int c  = blockIdx.x % NCHUNK;
    const int r0 = c * CHUNK;
    const int tid = threadIdx.x;

    const float* q_h   = qkv + (size_t)h * M_TOK * D_HEAD;
    const float* k_new = qkv + 1 * (H_HEADS * M_TOK * D_HEAD) + (size_t)h * M_TOK * D_HEAD;
    const float* v_new = qkv + 2 * (H_HEADS * M_TOK * D_HEAD) + (size_t)h * M_TOK * D_HEAD;

    const int wave = tid >> 5;
    const int lane = tid & 31;

    // TDM bulk copy of Q (2048 contiguous f32) into LDS; no VGPR round-trip.
    if (wave == 0) {
        tdm_load_row_f32((unsigned int)(uintptr_t)(void*)Qs,
                         (unsigned long long)(uintptr_t)q_h,
                         (unsigned int)(M_TOK * D_HEAD));
        __builtin_amdgcn_s_wait_tensorcnt(0);
    }
    for (int i = tid; i < M_TOK * CHUNK; i += 256) (&Ss[0][0])[i] = -1e30f;
    __syncthreads();

    const int row  = lane & 15;
    const int half = lane >> 4;
    const int col  = lane & 15;

    // ---- Stage 1: scores (wave w owns chunk rows [w*64, w*64+64)) ----
    for (int tt = 0; tt < 4; ++tt) {
        const int gr = r0 + wave * 64 + tt * 16;   // wave-uniform
        if (gr < S_TOTAL) {
            const int kr = gr + col;               // this lane's K row
            const float* krow = (kr < P_CACHE)
                ? (cacheK + ((size_t)h * P_CACHE + kr) * D_HEAD)
                : (k_new + (size_t)(kr - P_CACHE) * D_HEAD);
            v8f acc = {};
            #pragma unroll 8
            for (int k0 = 0; k0 < D_HEAD; k0 += 4) {
                v2f a = *(const v2f*)(Qs + row * D_HEAD + k0 + half * 2);
                v2f b = *(const v2f*)(krow + k0 + half * 2);
                acc = wmma_f32_16x16x4(a, b, acc);
            }
            const int sc = wave * 64 + tt * 16 + col;
            #pragma unroll
            for (int j = 0; j < 8; ++j) Ss[j + half * 8][sc] = acc[j];
        }
    }
    __syncthreads();

    // ---- Chunk softmax: 16 threads per row, 32 cols per thread ----
    const int srow = tid >> 4;
    const int tcol = tid & 15;
    float mx = -1e30f;
    for (int i = 0; i < 32; ++i) mx = fmaxf(mx, Ss[srow][tcol * 32 + i]);
    red[srow][tcol] = mx;
    __syncthreads();
    float m_row = -1e30f;
    for (int i = 0; i < 16; ++i) m_row = fmaxf(m_row, red[srow][i]);
    __syncthreads();

    float sum = 0.0f;
    for (int i = 0; i < 32; ++i) {
        float p = __expf(Ss[srow][tcol * 32 + i] - m_row);
        Ss[srow][tcol * 32 + i] = p;
        sum += p;
    }
    red[srow][tcol] = sum;
    __syncthreads();
    float l_row = 0.0f;
    for (int i = 0; i < 16; ++i) l_row += red[srow][i];
    if (tcol == 0) {
        stat_m[((size_t)h * NCHUNK + c) * M_TOK + srow] = m_row;
        stat_l[((size_t)h * NCHUNK + c) * M_TOK + srow] = l_row;
    }
    __syncthreads();

    // ---- Stage 2: O = P @ V (wave w owns D columns [w*16, w*16+16)) ----
    const int d0 = wave * 16;
    v8f acc = {};
    for (int k0 = 0; k0 < CHUNK; k0 += 4) {
        v2f a = *(const v2f*)(&Ss[row][k0 + half * 2]);
        int vr0 = r0 + k0 + half * 2;
        int vr1 = vr0 + 1;
        vr0 = (vr0 < S_TOTAL) ? vr0 : (S_TOTAL - 1);   // weight is 0 beyond S
        vr1 = (vr1 < S_TOTAL) ? vr1 : (S_TOTAL - 1);
        const float* vrow0 = (vr0 < P_CACHE)
            ? (cacheV + ((size_t)h * P_CACHE + vr0) * D_HEAD)
            : (v_new + (size_t)(vr0 - P_CACHE) * D_HEAD);
        const float* vrow1 = (vr1 < P_CACHE)
            ? (cacheV + ((size_t)h * P_CACHE + vr1) * D_HEAD)
            : (v_new + (size_t)(vr1 - P_CACHE) * D_HEAD);
        v2f b;
        b.x = vrow0[d0 + col];
        b.y = vrow1[d0 + col];
        acc = wmma_f32_16x16x4(a, b, acc);
    }
    #pragma unroll
    for (int j = 0; j < 8; ++j) {
        int r = j + half * 8;
        o_part[(((size_t)h * NCHUNK + c) * M_TOK + r) * D_HEAD + d0 + col] = acc[j];
    }
}

// ---------------------------------------------------------------------------
// Kernel 3: combine chunk partials with log-sum-exp; emit [M, H*D].
// ---------------------------------------------------------------------------
__global__ __launch_bounds__(256) void attn_reduce(
    const float* __restrict__ o_part,
    const float* __restrict__ stat_m,
    const float* __restrict__ stat_l,
    float* __restrict__ out) {

    const int g = blockIdx.x * 256 + threadIdx.x;    // 0..H*M*D-1
    const int h = g / (M_TOK * D_HEAD);
    const int rem = g % (M_TOK * D_HEAD);
    const int m = rem / D_HEAD;
    const int d = rem % D_HEAD;

    float gm = -1e30f;
    for (int c = 0; c < NCHUNK; ++c)
        gm = fmaxf(gm, stat_m[((size_t)h * NCHUNK + c) * M_TOK + m]);

    float num = 0.0f, den = 0.0f;
    for (int c = 0; c < NCHUNK; ++c) {
        float w = __expf(stat_m[((size_t)h * NCHUNK + c) * M_TOK + m] - gm);
        den += w * stat_l[((size_t)h * NCHUNK + c) * M_TOK + m];
        num += w * o_part[(((size_t)h * NCHUNK + c) * M_TOK + m) * D_HEAD + d];
    }
    out[(size_t)m * N_DIM + h * D_HEAD + d] = num / den;
}

// ---------------------------------------------------------------------------
extern "C" void kernel_launch(void* const* d_in, const int* in_sizes, int n_in,
                              void* d_out, int out_size, void* d_ws, size_t ws_size,
                              hipStream_t stream) {
    const float* X  = (const float*)d_in[0];
    const float* Wq = (const float*)d_in[1];
    const float* Wk = (const float*)d_in[2];
    const float* Wv = (const float*)d_in[3];
    const float* cK = (const float*)d_in[4];
    const float* cV = (const float*)d_in[5];
    float* out = (float*)d_out;

    // Workspace layout (floats): qkv | o_part | stat_m | stat_l  (~5.3 MB)
    float* ws     = (float*)d_ws;
    float* qkv    = ws;                                   // 196608
    float* o_part = qkv + QKV_ELEMS;                      // H*NCHUNK*M*D
    float* st_m   = o_part + (size_t)H_HEADS * NCHUNK * M_TOK * D_HEAD;
    float* st_l   = st_m + (size_t)H_HEADS * NCHUNK * M_TOK;

    zero_qkv<<<192, 256, 0, stream>>>(qkv);
    qkv_gemm<<<(3 * 256 * KSPLIT) / 8, 256, 0, stream>>>(X, Wq, Wk, Wv, qkv);
    attn_chunk<<<H_HEADS * NCHUNK, 256, 0, stream>>>(cK, cV, qkv, o_part, st_m, st_l);
    attn_reduce<<<(H_HEADS * M_TOK * D_HEAD) / 256, 256, 0, stream>>>(o_part, st_m, st_l, out);
}